// OLELoss_79620103733263
// MI455X (gfx1250) — compile-verified
//
#include <hip/hip_runtime.h>
#include <hip/hip_bf16.h>
#include <math.h>

typedef __attribute__((ext_vector_type(2))) float v2f;
typedef __attribute__((ext_vector_type(8))) float v8f;

#define DIMS   128
#define NTOT   8192
#define LAMBDA 0.25f
#define EPSV   1e-6f

// -------- kernel 0: zero the scalar output --------
__global__ void zero_out(float* out) {
    if (threadIdx.x == 0 && blockIdx.x == 0) out[0] = 0.0f;
}

// -------- kernel 1: per-row squared norms (one wave32 per row) --------
__global__ void row_sqnorms(const float* __restrict__ F, float* __restrict__ sn) {
    const int wave = threadIdx.x >> 5;
    const int lane = threadIdx.x & 31;
    const int row  = blockIdx.x * 8 + wave;           // 8 waves per block
    const float4* p = (const float4*)(F + (size_t)row * DIMS);
    float4 v = p[lane];                                // 32 lanes * 4 = 128
    float s = v.x * v.x + v.y * v.y + v.z * v.z + v.w * v.w;
    #pragma unroll
    for (int m = 16; m > 0; m >>= 1) s += __shfl_xor(s, m, 32);
    if (lane == 0) sn[row] = s;
}

// -------- kernel 2: fused Gram-matrix + OLE loss --------
// Block: 256 threads = 8 waves. Block tile: 128 rows (i) x 16 cols (j).
// Each wave computes one 16x16 tile of F*F^T with V_WMMA_F32_16X16X4_F32,
// then applies the loss epilogue and accumulates a pre-scaled partial sum.
__global__ void ole_main(const float* __restrict__ F,
                         const int*   __restrict__ labels,
                         const float* __restrict__ sn,
                         float* __restrict__ out) {
    __shared__ float Bs[16 * 132];        // 16 j-rows of features, padded stride
    __shared__ float wsum[8];

    const int tid  = threadIdx.x;
    const int wave = tid >> 5;
    const int lane = tid & 31;
    const int i0   = blockIdx.x * 128 + wave * 16;
    const int j0   = blockIdx.y * 16;

    // Stage the 16x128 B-strip (rows j0..j0+15 of F) into LDS.
    // 256 threads x 8 floats = 2048 floats.
    {
        const int r = tid >> 4;            // 0..15
        const int c = (tid & 15) * 8;      // 0..120 step 8
        const float4* src = (const float4*)(F + (size_t)(j0 + r) * DIMS + c);
        float4 v0 = src[0];
        float4 v1 = src[1];
        *(float4*)(&Bs[r * 132 + c])     = v0;
        *(float4*)(&Bs[r * 132 + c + 4]) = v1;
    }
    __syncthreads();

    // Fragment addressing per ISA layout (16x4 f32 A, 4x16 f32 B):
    //   lanes 0-15 : M/N = lane,    K-pair {k+0, k+1}
    //   lanes 16-31: M/N = lane-16, K-pair {k+2, k+3}
    const int half  = lane >> 4;          // 0 or 1
    const int mn    = lane & 15;
    const int koff  = half * 2;
    const int arow  = i0 + mn;

    v8f c = {};
    #pragma unroll 4
    for (int k = 0; k < DIMS; k += 4) {
        v2f a = *(const v2f*)(F + (size_t)arow * DIMS + k + koff);
        v2f b = *(const v2f*)(&Bs[mn * 132 + k + koff]);
        c = __builtin_amdgcn_wmma_f32_16x16x4_f32(
                /*neg_a=*/false, a, /*neg_b=*/false, b,
                /*c_mod=*/(short)0, c, /*reuse_a=*/false, /*reuse_b=*/false);
    }

    // Epilogue: C layout -> VGPR r holds (M = r + 8*half, N = mn) of the tile.
    const int jcol = j0 + mn;
    const int lj   = labels[jcol];
    const float snj = sn[jcol];
    float local = 0.0f;
    #pragma unroll
    for (int r = 0; r < 8; ++r) {
        const int irow = i0 + half * 8 + r;
        const float dot = c[r];
        float sq = sn[irow] + snj - 2.0f * dot;
        sq = fmaxf(sq, 0.0f);
        const float nrm = (sq > 0.0f) ? sqrtf(sq) : 0.0f;
        const bool matched = (labels[irow] == lj);
        local += matched ? nrm : (LAMBDA / (nrm + EPSV));
    }
    local *= 1.0f / ((float)NTOT * (float)NTOT);

    // wave reduce
    #pragma unroll
    for (int m = 16; m > 0; m >>= 1) local += __shfl_xor(local, m, 32);
    if (lane == 0) wsum[wave] = local;
    __syncthreads();

    if (tid == 0) {
        float s = 0.0f;
        #pragma unroll
        for (int w = 0; w < 8; ++w) s += wsum[w];
        atomicAdd(out, s);
    }
}

extern "C" void kernel_launch(void* const* d_in, const int* in_sizes, int n_in,
                              void* d_out, int out_size, void* d_ws, size_t ws_size,
                              hipStream_t stream) {
    const float* F      = (const float*)d_in[0];   // [8192,128] fp32
    const int*   labels = (const int*)d_in[1];     // [8192]
    float*       out    = (float*)d_out;           // scalar
    float*       sn     = (float*)d_ws;            // [8192] squared norms

    zero_out<<<1, 1, 0, stream>>>(out);
    row_sqnorms<<<NTOT / 8, 256, 0, stream>>>(F, sn);

    dim3 grid(NTOT / 128, NTOT / 16);              // 64 x 512 blocks
    ole_main<<<grid, 256, 0, stream>>>(F, labels, sn, out);
}